// RosaBitsLayer_80358838108512
// MI455X (gfx1250) — compile-verified
//
#include <hip/hip_runtime.h>
#include <cstdint>
#include <cstddef>

typedef __attribute__((ext_vector_type(2))) float v2f;
typedef __attribute__((ext_vector_type(4))) float v4f;
typedef __attribute__((ext_vector_type(8))) float v8f;

namespace {
constexpr int kB = 4, kT = 2048, kC = 2048, kD = 8, kH = 256;
constexpr int kChunk = 256;
constexpr int kNCh  = kT / kChunk;     // 8 chunks per sequence
constexpr int kL    = 16;              // microtile timesteps (WMMA M)
constexpr int kMT   = kChunk / kL;     // 16 microtiles per chunk
constexpr int kUnits = kB * kH * kNCh; // 8192 wave-units
constexpr int kWaves = 8, kTPB = kWaves * 32;
}

// ---- CDNA5 async global->LDS staging (ASYNCcnt), with sync fallback --------
#if __has_builtin(__builtin_amdgcn_global_load_async_to_lds_b128) && \
    __has_builtin(__builtin_amdgcn_s_wait_asynccnt)
#define ROSA_ASYNC 1
#else
#define ROSA_ASYNC 0
#endif

// Builtin signature (from hipcc diagnostic): param0 = generic `int
// __attribute__((vector_size(16)))*` (global src), param1 = addrspace(3)
// pointer (LDS dst), then imm offset + imm cpol.
typedef int v4i_vs __attribute__((vector_size(16)));
typedef __attribute__((address_space(3))) v4i_vs as3_v4i;
#define GLOBP(p) ((v4i_vs*)(p))
#define LDSP(p)  ((as3_v4i*)(unsigned int)(unsigned long long)(const void*)(p))

__device__ __forceinline__ void wait_async_le(int) {}
#if ROSA_ASYNC
#define WAIT_ASYNC(n) __builtin_amdgcn_s_wait_asynccnt(n)
#else
#define WAIT_ASYNC(n) wait_async_le(n)
#endif

// D = A(16x4 f32) * B(4x16 f32) + C(16x16 f32); full-f32 CDNA5 WMMA.
__device__ __forceinline__ v8f wmma_f32(v2f a, v2f b, v8f c) {
  return __builtin_amdgcn_wmma_f32_16x16x4_f32(false, a, false, b, (short)0, c,
                                               false, false);
}

// Stage one 16x8 f32 tile into a padded [16][16] LDS tile (lane pair = row).
__device__ __forceinline__ void stage_tile(const float* __restrict__ gbase,
                                           int t0, int lane, float* tile) {
  const int row = lane >> 1, cg = (lane & 1) * 4;
  const float* g = gbase + (size_t)(t0 + row) * kC + cg;
  float* l = tile + row * 16 + cg;
#if ROSA_ASYNC
  __builtin_amdgcn_global_load_async_to_lds_b128(GLOBP(g), LDSP(l), 0, 0);
#else
  *(v4f*)l = *(const v4f*)g;
#endif
}

// Zero the pad columns 8..15 of a [16][16] tile (done once; staging never
// touches them, so all B-operand reads can be unconditional).
__device__ __forceinline__ void zero_pad(float* tile, int lane) {
  const int row = lane >> 1, cg = 8 + (lane & 1) * 4;
  v4f z = {0.f, 0.f, 0.f, 0.f};
  *(v4f*)(tile + row * 16 + cg) = z;
}

// A-operand slice j of a row-major [16][16] tile: lane (ln,hi):
// {X[ln][4j+2hi], X[ln][4j+2hi+1]}  -> one ds_load_b64
__device__ __forceinline__ v2f read_astyle(const float* tile, int j, int ln, int hi) {
  return *(const v2f*)(tile + ln * 16 + j * 4 + hi * 2);
}

// B-operand slice j: lane (ln,hi), vgpr r: X[4j+2hi+r][ln].
// Pad columns are zero, so no predication -> ds_load_2addr_b32.
__device__ __forceinline__ v2f read_bstyle(const float* tile, int j, int ln, int hi) {
  const int k0 = j * 4 + hi * 2;
  v2f r;
  r.x = tile[(k0 + 0) * 16 + ln];
  r.y = tile[(k0 + 1) * 16 + ln];
  return r;
}

// ---------------- Pass 1: per-chunk partial states  S_c = sum_t k_t^T v_t ----
struct alignas(16) P1Lds {
  float k[2][16 * 16];
  float v[2][16 * 16];
};

__global__ void __launch_bounds__(kTPB)
rosa_partial(const float* __restrict__ xk, const float* __restrict__ xv,
             float* __restrict__ part) {
  __shared__ P1Lds lds[kWaves];
  const int tid = threadIdx.x, w = tid >> 5, lane = tid & 31;
  const int ln = lane & 15, hi = lane >> 4;
  const int unit = blockIdx.x * kWaves + w;
  const int ch = unit % kNCh, bh = unit / kNCh, h = bh % kH, b = bh / kH;
  const size_t base = (size_t)b * kT * kC + (size_t)h * kD;
  const float* kb = xk + base;
  const float* vb = xv + base;
  P1Lds& L = lds[w];

  zero_pad(L.k[0], lane); zero_pad(L.k[1], lane);
  zero_pad(L.v[0], lane); zero_pad(L.v[1], lane);

  const int tbase = ch * kChunk;
  stage_tile(kb, tbase, lane, L.k[0]);
  stage_tile(vb, tbase, lane, L.v[0]);

  v8f s = {0.f, 0.f, 0.f, 0.f, 0.f, 0.f, 0.f, 0.f};
  for (int mt = 0; mt < kMT; ++mt) {
    const int cur = mt & 1;
    if (mt + 1 < kMT) {  // software pipeline: issue next tile, wait for current
      stage_tile(kb, tbase + (mt + 1) * kL, lane, L.k[cur ^ 1]);
      stage_tile(vb, tbase + (mt + 1) * kL, lane, L.v[cur ^ 1]);
      WAIT_ASYNC(2);
    } else {
      WAIT_ASYNC(0);
    }
#pragma unroll
    for (int j = 0; j < 4; ++j) {              // K-dim = 16 timesteps
      v2f ak = read_bstyle(L.k[cur], j, ln, hi);  // A = K^T (rows>=8 zero-pad)
      v2f bv = read_bstyle(L.v[cur], j, ln, hi);  // B = V   (cols>=8 zero-pad)
      s = wmma_f32(ak, bv, s);
    }
  }
  // D-layout: element (m,n) at lane n (hi==0 for m<8), vgpr m.
  if (hi == 0 && ln < 8) {
    float* p = part + (size_t)unit * 64;
#pragma unroll
    for (int m = 0; m < 8; ++m) p[m * 8 + ln] = s[m];
  }
}

// ---------------- Pass 2: exclusive prefix over chunks per (b,h) -------------
__global__ void rosa_scan(const float* __restrict__ part, float* __restrict__ pref) {
  const int idx = blockIdx.x * blockDim.x + threadIdx.x;  // kB*kH*64 threads
  const int e = idx & 63, bh = idx >> 6;
  float run = 0.0f;
#pragma unroll
  for (int c = 0; c < kNCh; ++c) {
    const size_t o = ((size_t)bh * kNCh + c) * 64 + e;
    pref[o] = run;
    run += part[o];
  }
}

// ---------------- Pass 3: outputs  O = Q*S0 + mask(QK^T)*V, chain state ------
struct alignas(16) P3Lds {
  float q[2][16 * 16];
  float k[2][16 * 16];
  float v[2][16 * 16];
  float am[16 * 16];  // masked QK^T transpose buffer; reused for O transpose
  float s0[16 * 16];  // running state staged for B-operand reads (rows 0..7)
};

__global__ void __launch_bounds__(kTPB)
rosa_output(const float* __restrict__ xq, const float* __restrict__ xk,
            const float* __restrict__ xv, const float* __restrict__ emb,
            const float* __restrict__ pref, float* __restrict__ out) {
  __shared__ P3Lds lds[kWaves];
  const int tid = threadIdx.x, w = tid >> 5, lane = tid & 31;
  const int ln = lane & 15, hi = lane >> 4;
  const int unit = blockIdx.x * kWaves + w;
  const int ch = unit % kNCh, bh = unit / kNCh, h = bh % kH, b = bh / kH;
  const size_t base = (size_t)b * kT * kC + (size_t)h * kD;
  const float* qb = xq + base;
  const float* kb = xk + base;
  const float* vb = xv + base;
  float* ob = out + base;
  const float embv = (ln < 8) ? emb[h * kD + ln] : 0.0f;
  P3Lds& L = lds[w];

  zero_pad(L.q[0], lane); zero_pad(L.q[1], lane);
  zero_pad(L.k[0], lane); zero_pad(L.k[1], lane);
  zero_pad(L.v[0], lane); zero_pad(L.v[1], lane);

  // Chunk-start state S0 (exclusive prefix) into D-layout accumulator.
  v8f sAcc = {0.f, 0.f, 0.f, 0.f, 0.f, 0.f, 0.f, 0.f};
  {
    const float* p = pref + (size_t)unit * 64;
    if (hi == 0 && ln < 8) {
#pragma unroll
      for (int m = 0; m < 8; ++m) sAcc[m] = p[m * 8 + ln];
    }
  }

  const int tbase = ch * kChunk;
  stage_tile(qb, tbase, lane, L.q[0]);
  stage_tile(kb, tbase, lane, L.k[0]);
  stage_tile(vb, tbase, lane, L.v[0]);

  for (int mt = 0; mt < kMT; ++mt) {
    const int cur = mt & 1;
    const int t0 = tbase + mt * kL;
    if (mt + 1 < kMT) {
      stage_tile(qb, t0 + kL, lane, L.q[cur ^ 1]);
      stage_tile(kb, t0 + kL, lane, L.k[cur ^ 1]);
      stage_tile(vb, t0 + kL, lane, L.v[cur ^ 1]);
      WAIT_ASYNC(3);
    } else {
      WAIT_ASYNC(0);
    }

    // Stage state-before-this-tile (masked VALUES, unconditional stores:
    // pad cols and rows>=8 get zeros, so reads need no predication).
#pragma unroll
    for (int vv = 0; vv < 8; ++vv) {
      const int m = vv + 8 * hi;
      L.s0[m * 16 + ln] = (hi == 0 && ln < 8) ? sAcc[vv] : 0.0f;
    }

    // S = Q K^T (16x16, K-dim 8). K loaded A-style == K^T as B operand.
    const v2f aq0 = read_astyle(L.q[cur], 0, ln, hi);
    const v2f aq1 = read_astyle(L.q[cur], 1, ln, hi);
    const v2f bk0 = read_astyle(L.k[cur], 0, ln, hi);
    const v2f bk1 = read_astyle(L.k[cur], 1, ln, hi);
    v8f sc = {0.f, 0.f, 0.f, 0.f, 0.f, 0.f, 0.f, 0.f};
    sc = wmma_f32(aq0, bk0, sc);
    sc = wmma_f32(aq1, bk1, sc);

    // Causal-inclusive mask (keep n<=m) + D-layout -> A-layout via LDS.
#pragma unroll
    for (int vv = 0; vv < 8; ++vv) {
      const int m = vv + 8 * hi;
      L.am[m * 16 + ln] = (ln <= m) ? sc[vv] : 0.0f;
    }

    // O = Q @ S0 (inter-tile)  +  mask(QK^T) @ V (intra-tile, K-dim 16);
    // state += K^T V chained through the WMMA C operand.
    v8f o = {0.f, 0.f, 0.f, 0.f, 0.f, 0.f, 0.f, 0.f};
    o = wmma_f32(aq0, read_bstyle(L.s0, 0, ln, hi), o);
    o = wmma_f32(aq1, read_bstyle(L.s0, 1, ln, hi), o);
#pragma unroll
    for (int j = 0; j < 4; ++j) {
      const v2f am = read_astyle(L.am, j, ln, hi);
      const v2f bv = read_bstyle(L.v[cur], j, ln, hi);
      o = wmma_f32(am, bv, o);
      const v2f ak = read_bstyle(L.k[cur], j, ln, hi);
      sAcc = wmma_f32(ak, bv, sAcc);
    }

    // O tile: D-layout -> row-major via LDS (reuse am), then b128 stores.
#pragma unroll
    for (int vv = 0; vv < 8; ++vv) {
      const int m = vv + 8 * hi;
      L.am[m * 16 + ln] = (ln < 8) ? o[vv] * embv : 0.0f;
    }
    {
      const int row = lane >> 1, cg = (lane & 1) * 4;
      v4f ov = *(const v4f*)(L.am + row * 16 + cg);
      *(v4f*)(ob + (size_t)(t0 + row) * kC + cg) = ov;
    }
  }
}

extern "C" void kernel_launch(void* const* d_in, const int* in_sizes, int n_in,
                              void* d_out, int out_size, void* d_ws, size_t ws_size,
                              hipStream_t stream) {
  const float* xq  = (const float*)d_in[0];
  const float* xk  = (const float*)d_in[1];
  const float* xv  = (const float*)d_in[2];
  const float* emb = (const float*)d_in[3];
  float* out = (float*)d_out;

  float* part = (float*)d_ws;                      // kUnits*64 f32 = 2 MiB
  float* pref = part + (size_t)kUnits * 64;        // kUnits*64 f32 = 2 MiB

  rosa_partial<<<kUnits / kWaves, kTPB, 0, stream>>>(xk, xv, part);
  rosa_scan<<<(kB * kH * 64) / 256, 256, 0, stream>>>(part, pref);
  rosa_output<<<kUnits / kWaves, kTPB, 0, stream>>>(xq, xk, xv, emb, pref, out);
}